// EnhancedVectorQuantizer_75445395522044
// MI455X (gfx1250) — compile-verified
//
#include <hip/hip_runtime.h>
#include <hip/hip_bf16.h>

// ---- problem constants -----------------------------------------------------
#define B_ROWS      65536
#define D_DIM       64
#define K_CODES     16
#define N_ELEMS     (B_ROWS * D_DIM)        // 4194304 f32
#define NV4         (N_ELEMS / 4)           // 1048576 float4
#define BLOCKS      2048
#define THREADS     256
#define TOT_THREADS (BLOCKS * THREADS)      // 524288 (multiple of 16 -> fixed d-group/thread)
#define ITERS       (NV4 / TOT_THREADS)     // 2, exact cover, no bounds checks

typedef __attribute__((ext_vector_type(4))) unsigned int v4u_t;
typedef __attribute__((ext_vector_type(8))) int          v8i_t;
typedef __attribute__((ext_vector_type(4))) int          v4i_t;

// ---------------------------------------------------------------------------
// Kernel 1: stream z, per-scalar 16-way argmin against LDS/VGPR-staged
// codebook, write z_q + indices, accumulate per-block squared-error partials.
// ---------------------------------------------------------------------------
__global__ void __launch_bounds__(THREADS)
vq_main(const float* __restrict__ z, const float* __restrict__ cbw,
        float* __restrict__ out, float* __restrict__ partial)
{
    __shared__ float lds_cb[D_DIM * K_CODES];   // 4 KB codebook stage
    __shared__ float red[THREADS];              // loss reduction tree

    const int t = threadIdx.x;

    // ---- stage codebook (64x16 f32, row-major) into LDS via the TDM -------
#if defined(__gfx1250__) && __has_builtin(__builtin_amdgcn_tensor_load_to_lds)
    {
        // D# group 0: count=1, lds_addr, global_addr[56:0], type=2
        unsigned long long ga = (unsigned long long)(size_t)cbw;
        unsigned lds_off = (unsigned)(unsigned long long)(size_t)&lds_cb[0]; // flat->LDS low 32
        v4u_t g0 = { 1u,                                              // count=1, user mode
                     lds_off,                                         // bits 63:32
                     (unsigned)(ga & 0xFFFFFFFFull),                  // bits 95:64
                     (unsigned)((ga >> 32) & 0x01FFFFFFull) | (2u << 30) }; // addr hi + type=2

        // D# group 1: data_size=4B, tensor_dim0=1024, tensor_dim1=1,
        //             tile_dim0=1024, tile_dim1=1, dim0_stride=1024
        v8i_t g1 = { (int)(2u << 16),          // wg_mask=0, data_size=2 (4B)
                     (int)(1024u << 16),       // tensor_dim0[15:0] << 16
                     (int)(1u << 16),          // tensor_dim0 hi=0 | tensor_dim1[15:0]=1
                     (int)(1024u << 16),       // tensor_dim1 hi=0 | tile_dim0=1024
                     1,                        // tile_dim1=1, tile_dim2=0
                     1024,                     // tensor_dim0_stride[31:0]
                     (int)(1024u << 16),       // d0_stride hi=0 | tensor_dim1_stride lo
                     0 };
        v4i_t g2 = { 0, 0, 0, 0 };
        v4i_t g3 = { 0, 0, 0, 0 };
#if __clang_major__ >= 23
        v8i_t gz = { 0, 0, 0, 0, 0, 0, 0, 0 };
        __builtin_amdgcn_tensor_load_to_lds(g0, g1, g2, g3, gz, 0);
#else
        __builtin_amdgcn_tensor_load_to_lds(g0, g1, g2, g3, 0);
#endif
        __builtin_amdgcn_s_wait_tensorcnt(0);
    }
#else
    for (int i = t; i < D_DIM * K_CODES; i += THREADS) lds_cb[i] = cbw[i];
#endif
    __syncthreads();

    // ---- pull this thread's 4 codebook rows into VGPRs --------------------
    // grid stride (524288) is a multiple of 16 float4s => d-group is
    // loop-invariant per thread: d0 = (t & 15) * 4.
    const int d0 = (t & 15) * 4;
    float cb[4][K_CODES];
#pragma unroll
    for (int j = 0; j < 4; ++j) {
        const float4* r = (const float4*)&lds_cb[(d0 + j) * K_CODES];
#pragma unroll
        for (int q = 0; q < 4; ++q) {
            float4 c4 = r[q];                  // ds_load
            cb[j][q * 4 + 0] = c4.x;
            cb[j][q * 4 + 1] = c4.y;
            cb[j][q * 4 + 2] = c4.z;
            cb[j][q * 4 + 3] = c4.w;
        }
    }

    const float4* z4      = (const float4*)z;
    float4*       zq4     = (float4*)out;
    float*        idx_out = out + N_ELEMS + 1;   // [z_q | loss | indices]

    float acc = 0.0f;
    const unsigned v0 = (unsigned)blockIdx.x * THREADS + t;

    // Issue all global loads up front (2-deep MLP per thread), then compute.
    float4 zin[ITERS];
#pragma unroll
    for (int it = 0; it < ITERS; ++it) {
        unsigned v = v0 + (unsigned)it * TOT_THREADS;
        __builtin_prefetch(&z4[v], 0, 0);      // global_prefetch_b8 (cacheline pull)
        zin[it] = z4[v];                       // global_load_b128
    }

#pragma unroll
    for (int it = 0; it < ITERS; ++it) {
        const unsigned v = v0 + (unsigned)it * TOT_THREADS;
        float zval[4] = { zin[it].x, zin[it].y, zin[it].z, zin[it].w };
        float zq_[4], id_[4];

#pragma unroll
        for (int j = 0; j < 4; ++j) {
            float zv   = zval[j];
            float c0   = cb[j][0];
            float df0  = zv - c0;
            float best = df0 * df0;
            float bval = c0;
            float bidx = 0.0f;
#pragma unroll
            for (int k = 1; k < K_CODES; ++k) {
                float c  = cb[j][k];
                float df = zv - c;
                float d2 = df * df;
                bool  m  = d2 < best;          // strict '<' => first-min, matches argmin
                best = m ? d2 : best;
                bval = m ? c  : bval;
                bidx = m ? (float)k : bidx;
            }
            acc   += best;                     // == (z - z_q)^2 for chosen code
            zq_[j] = bval;
            id_[j] = bidx;
        }

        zq4[v] = make_float4(zq_[0], zq_[1], zq_[2], zq_[3]); // global_store_b128
        float* ip = idx_out + (size_t)v * 4;   // region base misaligned by 4B -> b32 stores
        ip[0] = id_[0]; ip[1] = id_[1]; ip[2] = id_[2]; ip[3] = id_[3];
    }

    // ---- deterministic fixed-order block reduction ------------------------
    red[t] = acc;
#pragma unroll
    for (int off = THREADS / 2; off > 0; off >>= 1) {
        __syncthreads();
        if (t < off) red[t] += red[t + off];
    }
    if (t == 0) partial[blockIdx.x] = red[0];
}

// ---------------------------------------------------------------------------
// Kernel 2: deterministic final reduction of 2048 block partials -> vq_loss.
// vq_loss = S * (1/(B*D) + 0.25/B)
// ---------------------------------------------------------------------------
__global__ void __launch_bounds__(256)
vq_finalize(const float* __restrict__ partial, float* __restrict__ out)
{
    __shared__ float red[256];
    const int t = threadIdx.x;
    float s = 0.0f;
#pragma unroll
    for (int q = 0; q < BLOCKS / 256; ++q)      // fixed order: deterministic
        s += partial[q * 256 + t];
    red[t] = s;
#pragma unroll
    for (int off = 128; off > 0; off >>= 1) {
        __syncthreads();
        if (t < off) red[t] += red[t + off];
    }
    if (t == 0) {
        const double scale = 1.0 / ((double)B_ROWS * (double)D_DIM)
                           + 0.25 / (double)B_ROWS;
        out[N_ELEMS] = (float)((double)red[0] * scale);
    }
}

// ---------------------------------------------------------------------------
extern "C" void kernel_launch(void* const* d_in, const int* in_sizes, int n_in,
                              void* d_out, int out_size, void* d_ws, size_t ws_size,
                              hipStream_t stream)
{
    (void)in_sizes; (void)n_in; (void)out_size; (void)ws_size;
    const float* z   = (const float*)d_in[0];   // (65536, 64) f32
    const float* cbw = (const float*)d_in[1];   // (64, 16) f32
    float* out     = (float*)d_out;             // [z_q (4M) | loss (1) | indices (4M)]
    float* partial = (float*)d_ws;              // 2048 f32 block partials (8 KB)

    vq_main<<<BLOCKS, THREADS, 0, stream>>>(z, cbw, out, partial);
    vq_finalize<<<1, 256, 0, stream>>>(partial, out);
}